// CustomMLPLayer_67877663146440
// MI455X (gfx1250) — compile-verified
//
#include <hip/hip_runtime.h>

// Gather-GEMM for sparse-MLP decode step on gfx1250 (MI455X, wave32).
// out[b, n] = sum_d x[b, d] * weight[indices[n], d]
//   x:       [64, 1, 4096]  fp32
//   weight:  [11008, 4096]  fp32
//   indices: [4403]         int32
//   out:     [64, 1, 4403]  fp32
//
// Strategy: V_WMMA_F32_16X16X4_F32 (fp32 in / fp32 acc, matches reference
// numerics). One wave per 16-neuron N-tile; the wave keeps all 4 M-tiles
// (M=64) in 4 accumulators so each gathered B row chunk is loaded once and
// reused by 4 WMMAs. Whole working set (~74MB) fits in the 192MB L2.

#define D_MODEL 4096
#define NEURONS 4403
#define BATCH   64

typedef __attribute__((ext_vector_type(2))) float v2f;
typedef __attribute__((ext_vector_type(8))) float v8f;

__global__ __launch_bounds__(32)
void gather_gemm_wmma_f32(const float* __restrict__ x,       // [64, 4096]
                          const float* __restrict__ weight,  // [11008, 4096]
                          const int*   __restrict__ indices, // [4403]
                          float* __restrict__ out)           // [64, 4403]
{
    const int lane  = threadIdx.x & 31;
    const int nl    = lane & 15;          // N (for B/D) or M (for A) within tile
    const int khi   = (lane >> 4) << 1;   // lanes 16-31 carry K=+2,+3 halves
    const int n0    = blockIdx.x * 16;
    const int n     = n0 + nl;

    // Branchless clamp keeps EXEC all-1s (WMMA requirement) through the loop.
    const int nc    = (n < NEURONS) ? n : (NEURONS - 1);
    const int row   = indices[nc];

    const float* __restrict__ bp  = weight + (size_t)row * D_MODEL + khi;
    const float* __restrict__ a0p = x + (size_t)nl * D_MODEL + khi;        // M  0..15
    const float* __restrict__ a1p = a0p + (size_t)16 * D_MODEL;            // M 16..31
    const float* __restrict__ a2p = a0p + (size_t)32 * D_MODEL;            // M 32..47
    const float* __restrict__ a3p = a0p + (size_t)48 * D_MODEL;            // M 48..63

    // Pull the gathered row toward the caches early.
    __builtin_prefetch(bp, 0, 1);

    v8f acc0 = {};
    v8f acc1 = {};
    v8f acc2 = {};
    v8f acc3 = {};

    #pragma unroll 4
    for (int k = 0; k < D_MODEL; k += 4) {
        // Per ISA 32-bit A 16x4 layout: lane<16 -> {K=k,k+1}, lane>=16 -> {K=k+2,k+3}
        // One b64 load each; B mirrors with lane = N of the gathered row.
        const v2f bv  = *(const v2f*)(bp  + k);
        const v2f av0 = *(const v2f*)(a0p + k);
        const v2f av1 = *(const v2f*)(a1p + k);
        const v2f av2 = *(const v2f*)(a2p + k);
        const v2f av3 = *(const v2f*)(a3p + k);

        // D = A x B + C ; (neg_a, A, neg_b, B, c_mod, C, reuse_a, reuse_b)
        acc0 = __builtin_amdgcn_wmma_f32_16x16x4_f32(false, av0, false, bv,
                                                     (short)0, acc0, false, false);
        acc1 = __builtin_amdgcn_wmma_f32_16x16x4_f32(false, av1, false, bv,
                                                     (short)0, acc1, false, false);
        acc2 = __builtin_amdgcn_wmma_f32_16x16x4_f32(false, av2, false, bv,
                                                     (short)0, acc2, false, false);
        acc3 = __builtin_amdgcn_wmma_f32_16x16x4_f32(false, av3, false, bv,
                                                     (short)0, acc3, false, false);
    }

    // D layout: VGPR r, lane L -> M = r + 8*(L>=16), N = L&15.
    if (n < NEURONS) {
        const int mhi = (lane >> 4) * 8;
        #pragma unroll
        for (int r = 0; r < 8; ++r) {
            const int m = mhi + r;
            out[(size_t)(m +  0) * NEURONS + n] = acc0[r];
            out[(size_t)(m + 16) * NEURONS + n] = acc1[r];
            out[(size_t)(m + 32) * NEURONS + n] = acc2[r];
            out[(size_t)(m + 48) * NEURONS + n] = acc3[r];
        }
    }
}

extern "C" void kernel_launch(void* const* d_in, const int* in_sizes, int n_in,
                              void* d_out, int out_size, void* d_ws, size_t ws_size,
                              hipStream_t stream) {
    const float* x       = (const float*)d_in[0];   // 64*1*4096 fp32
    const float* weight  = (const float*)d_in[1];   // 11008*4096 fp32
    const int*   indices = (const int*)d_in[2];     // 4403 int32
    float*       out     = (float*)d_out;           // 64*1*4403 fp32

    const int n_tiles = (NEURONS + 15) / 16;        // 276 waves, one per N-tile
    gather_gemm_wmma_f32<<<dim3(n_tiles), dim3(32), 0, stream>>>(x, weight, indices, out);
}